// StnModule_2791728742488
// MI455X (gfx1250) — compile-verified
//
#include <hip/hip_runtime.h>
#include <hip/hip_bf16.h>

typedef __attribute__((ext_vector_type(2))) float v2f;
typedef __attribute__((ext_vector_type(8))) float v8f;

#define KTOT   37632   // 3*112*112
#define NCOLS  20
#define KBLK   4       // K split across blocks
#define KCHNK  8       // K split across waves within a block
#define KPERW  (KTOT / KBLK / KCHNK)   // 1176 = 98 * 12
#define UNROLL_K 12
#define NITER  (KPERW / UNROLL_K)      // 98

// ---------------------------------------------------------------------------
// Kernel 1: partial GEMM  X[512,37632] @ W1[37632,20] via V_WMMA_F32_16X16X4_F32
// grid: (32 mTiles, 4 kBlocks), block: 256 threads (8 waves)
// wave w = kChunk (0..7).  Each wave computes BOTH N-tiles (cols 0-15 and the
// padded 16-31) from a single set of A loads, halving image traffic.
// Columns 20..31 load a clamped in-bounds duplicate of column 19: WMMA columns
// are independent dot products and those D columns are never stored, so the
// inner loop needs no masking and no branches.
// writes raw partial sums (no bias/tanh) to part[kb][512][20]
// ---------------------------------------------------------------------------
__global__ __launch_bounds__(256)
void stn_gemm1_wmma(const float* __restrict__ X,
                    const float* __restrict__ W1,
                    float* __restrict__ part) {
    __shared__ float red[8 * 32 * 16];   // 16 KB: [wave][lane][tile0 r0..7 | tile1 r0..7]

    const int tid   = threadIdx.x;
    const int wave  = tid >> 5;          // = kChunk, 0..7
    const int lane  = tid & 31;
    const int half  = lane >> 4;         // 0: lanes 0-15, 1: lanes 16-31
    const int sub   = lane & 15;

    const int mTile  = blockIdx.x;       // 0..31
    const int kb     = blockIdx.y;       // 0..3

    const int m      = mTile * 16 + sub;            // A row this lane feeds
    const int ncol0  = sub;                         // tile0 col (<16, always valid)
    const int ncol1  = (16 + sub < NCOLS) ? (16 + sub) : (NCOLS - 1); // clamp
    const int klo    = half * 2;                    // K sub-offset inside 4-step

    const int kStart = kb * (KTOT / KBLK) + wave * KPERW;

    const float* __restrict__ pA  = X  + (size_t)m * KTOT + kStart + klo;
    const float* __restrict__ pB0 = W1 + (size_t)(kStart + klo) * NCOLS + ncol0;
    const float* __restrict__ pB1 = W1 + (size_t)(kStart + klo) * NCOLS + ncol1;

    v8f acc00 = {}, acc01 = {}, acc02 = {};   // N-tile 0
    v8f acc10 = {}, acc11 = {}, acc12 = {};   // N-tile 1

    for (int it = 0; it < NITER; ++it) {
        // K = base..base+3 / +4..+7 / +8..+11   (this lane covers klo, klo+1)
        v2f a0 = *(const v2f*)(pA);
        v2f a1 = *(const v2f*)(pA + 4);
        v2f a2 = *(const v2f*)(pA + 8);

        v2f b00; b00.x = pB0[0];           b00.y = pB0[NCOLS];
        v2f b01; b01.x = pB0[4 * NCOLS];   b01.y = pB0[5 * NCOLS];
        v2f b02; b02.x = pB0[8 * NCOLS];   b02.y = pB0[9 * NCOLS];
        v2f b10; b10.x = pB1[0];           b10.y = pB1[NCOLS];
        v2f b11; b11.x = pB1[4 * NCOLS];   b11.y = pB1[5 * NCOLS];
        v2f b12; b12.x = pB1[8 * NCOLS];   b12.y = pB1[9 * NCOLS];

        acc00 = __builtin_amdgcn_wmma_f32_16x16x4_f32(
                    false, a0, false, b00, (short)0, acc00, false, false);
        acc10 = __builtin_amdgcn_wmma_f32_16x16x4_f32(
                    false, a0, false, b10, (short)0, acc10, false, false);
        acc01 = __builtin_amdgcn_wmma_f32_16x16x4_f32(
                    false, a1, false, b01, (short)0, acc01, false, false);
        acc11 = __builtin_amdgcn_wmma_f32_16x16x4_f32(
                    false, a1, false, b11, (short)0, acc11, false, false);
        acc02 = __builtin_amdgcn_wmma_f32_16x16x4_f32(
                    false, a2, false, b02, (short)0, acc02, false, false);
        acc12 = __builtin_amdgcn_wmma_f32_16x16x4_f32(
                    false, a2, false, b12, (short)0, acc12, false, false);

        pA  += UNROLL_K;
        pB0 += UNROLL_K * NCOLS;
        pB1 += UNROLL_K * NCOLS;
    }

    // stash per-wave accumulators in LDS
    #pragma unroll
    for (int r = 0; r < 8; ++r) {
        red[(wave * 32 + lane) * 16 + r]     = acc00[r] + acc01[r] + acc02[r];
        red[(wave * 32 + lane) * 16 + 8 + r] = acc10[r] + acc11[r] + acc12[r];
    }
    __syncthreads();

    // waves 0 (tile 0) and 1 (tile 1) reduce across the 8 K-chunk waves
    if (wave < 2) {
        const int nT = wave;
        #pragma unroll
        for (int r = 0; r < 8; ++r) {
            float s = 0.0f;
            #pragma unroll
            for (int kc = 0; kc < 8; ++kc)
                s += red[(kc * 32 + lane) * 16 + nT * 8 + r];
            const int row = r + half * 8;          // D layout: vgpr r, halves
            const int col = nT * 16 + sub;
            if (col < NCOLS)
                part[((size_t)kb * 512 + mTile * 16 + row) * NCOLS + col] = s;
        }
    }
}

// ---------------------------------------------------------------------------
// Kernel 2: h = tanh(sum_kb part + b1); theta = tanh(h @ W2 + b2)  -> [512,6]
// ---------------------------------------------------------------------------
__global__ void stn_theta(const float* __restrict__ part,
                          const float* __restrict__ b1,
                          const float* __restrict__ W2,
                          const float* __restrict__ b2,
                          float* __restrict__ theta) {
    const int idx = blockIdx.x * blockDim.x + threadIdx.x;
    if (idx >= 512 * 6) return;
    const int b = idx / 6;
    const int j = idx % 6;

    float acc = b2[j];
    #pragma unroll
    for (int kk = 0; kk < NCOLS; ++kk) {
        float s = b1[kk];
        #pragma unroll
        for (int kb = 0; kb < KBLK; ++kb)
            s += part[((size_t)kb * 512 + b) * NCOLS + kk];
        acc += tanhf(s) * W2[kk * 6 + j];
    }
    theta[idx] = tanhf(acc);
}

// ---------------------------------------------------------------------------
// Kernel 3: affine_grid (align_corners=False) + bilinear grid_sample (zeros)
// grid: (49, 512), block 256 -> one thread per pixel, all 3 channels
// ---------------------------------------------------------------------------
__global__ __launch_bounds__(256)
void stn_sample(const float* __restrict__ img,
                const float* __restrict__ theta,
                float* __restrict__ out) {
    const int b   = blockIdx.y;
    const int pix = blockIdx.x * 256 + threadIdx.x;   // 49*256 == 12544
    const int y   = pix / 112;
    const int x   = pix % 112;

    const float* __restrict__ t = theta + b * 6;
    const float t0 = t[0], t1 = t[1], t2 = t[2];
    const float t3 = t[3], t4 = t[4], t5 = t[5];

    const float gx = (2.0f * (float)x + 1.0f) * (1.0f / 112.0f) - 1.0f;
    const float gy = (2.0f * (float)y + 1.0f) * (1.0f / 112.0f) - 1.0f;

    const float sx = t0 * gx + t1 * gy + t2;
    const float sy = t3 * gx + t4 * gy + t5;

    const float ix = ((sx + 1.0f) * 112.0f - 1.0f) * 0.5f;
    const float iy = ((sy + 1.0f) * 112.0f - 1.0f) * 0.5f;

    const float x0f = floorf(ix), y0f = floorf(iy);
    const float wx1 = ix - x0f,  wx0 = 1.0f - wx1;
    const float wy1 = iy - y0f,  wy0 = 1.0f - wy1;
    const int x0 = (int)x0f, y0i = (int)y0f;
    const int x1 = x0 + 1,   y1i = y0i + 1;

    const float* __restrict__ base = img + (size_t)b * 3 * 12544;
    float a0 = 0.0f, a1 = 0.0f, a2 = 0.0f;

    #pragma unroll
    for (int corner = 0; corner < 4; ++corner) {
        const int xs = (corner & 1) ? x1 : x0;
        const int ys = (corner & 2) ? y1i : y0i;
        const float w = ((corner & 1) ? wx1 : wx0) * ((corner & 2) ? wy1 : wy0);
        if (xs >= 0 && xs < 112 && ys >= 0 && ys < 112) {
            const int off = ys * 112 + xs;
            a0 += w * base[off];
            a1 += w * base[12544 + off];
            a2 += w * base[2 * 12544 + off];
        }
    }

    float* __restrict__ o = out + (size_t)b * 3 * 12544 + pix;
    o[0]         = a0;
    o[12544]     = a1;
    o[2 * 12544] = a2;
}

// ---------------------------------------------------------------------------
extern "C" void kernel_launch(void* const* d_in, const int* in_sizes, int n_in,
                              void* d_out, int out_size, void* d_ws, size_t ws_size,
                              hipStream_t stream) {
    const float* img = (const float*)d_in[0];   // [512,3,112,112]
    const float* W1  = (const float*)d_in[1];   // [37632,20]
    const float* b1  = (const float*)d_in[2];   // [20]
    const float* W2  = (const float*)d_in[3];   // [20,6]
    const float* b2  = (const float*)d_in[4];   // [6]
    float* out = (float*)d_out;

    float* part  = (float*)d_ws;                // 4*512*20 floats
    float* theta = part + KBLK * 512 * NCOLS;   // 512*6 floats

    stn_gemm1_wmma<<<dim3(32, KBLK), 256, 0, stream>>>(img, W1, part);
    stn_theta<<<(512 * 6 + 255) / 256, 256, 0, stream>>>(part, b1, W2, b2, theta);
    stn_sample<<<dim3(49, 512), 256, 0, stream>>>(img, theta, out);
}